// CenterLoss_86672440033822
// MI455X (gfx1250) — compile-verified
//
#include <hip/hip_runtime.h>
#include <math.h>

// CDNA5 / gfx1250: wave32. WMMA 16x16x4 f32: A,B = 2 VGPRs (v2f), C/D = 8 VGPRs (v8f).
typedef __attribute__((ext_vector_type(2))) float v2f;
typedef __attribute__((ext_vector_type(8))) float v8f;

#define BATCH   256
#define FEAT    512
#define THREADS 256   // 8 wave32 waves per block
#define WAVES   8

// Full wave32 sum of (x + y) across all lanes using one V_WMMA_F32_16X16X4_F32.
// A = per-lane {x, y} (any slot mapping covers each tile element once),
// B = all-ones  =>  C[i][j] = rowsum_i for every column j.
// Per ISA §7.12.2, C VGPR r holds M=r (lanes 0-15) and M=r+8 (lanes 16-31), so
// h = Σ_r C[r] gives each half-wave one half of the total; one xor-16 finishes.
// Requires EXEC all-ones (uniform control flow at call sites).
__device__ __forceinline__ float wmma_wave_sum(float x, float y) {
    v2f a; a.x = x;    a.y = y;
    v2f b; b.x = 1.0f; b.y = 1.0f;
    v8f acc = {};
    acc = __builtin_amdgcn_wmma_f32_16x16x4_f32(
        /*neg_a=*/false, a, /*neg_b=*/false, b,
        /*c_mod=*/(short)0, acc, /*reuse_a=*/false, /*reuse_b=*/false);
    float h = acc[0] + acc[1] + acc[2] + acc[3] +
              acc[4] + acc[5] + acc[6] + acc[7];
    return h + __shfl_xor(h, 16, 32);   // every lane: full wave sum
}

__device__ __forceinline__ float wave_max(float v) {
#pragma unroll
    for (int m = 16; m >= 1; m >>= 1) v = fmaxf(v, __shfl_xor(v, m, 32));
    return v;
}

// Kernel A: one block per batch row.
//   ws_dist[row] = sum_d (feats[row,d] - centers[labels[row],d])^2
//   ws_lse[row]  = logsumexp(feats[row,:])
__global__ __launch_bounds__(THREADS)
void center_rowstats(const float* __restrict__ feats,
                     const float* __restrict__ centers,
                     const int*   __restrict__ labels,
                     float* __restrict__ ws_dist,
                     float* __restrict__ ws_lse) {
    const int row  = blockIdx.x;
    const int tid  = threadIdx.x;
    const int wave = tid >> 5;
    const int lane = tid & 31;

    const float* f = feats + (size_t)row * FEAT;
    const float* c = centers + (size_t)labels[row] * FEAT;

    // Each wave owns 64 contiguous columns; 2 coalesced elements per lane.
    const int d0 = wave * 64 + lane;
    const int d1 = d0 + 32;
    const float f0 = f[d0], f1 = f[d1];
    const float c0 = c[d0], c1 = c[d1];
    const float e0 = f0 - c0, e1 = f1 - c1;

    // Phase 1: wave-level Σ (f-c)^2 via WMMA, then cross-wave via LDS.
    const float wave_sq = wmma_wave_sum(e0 * e0, e1 * e1);

    __shared__ float sh[WAVES];
    __shared__ float sh_bcast;

    if (lane == 0) sh[wave] = wave_sq;
    __syncthreads();
    if (tid == 0) {
        float t = 0.0f;
#pragma unroll
        for (int w = 0; w < WAVES; ++w) t += sh[w];
        ws_dist[row] = t;
    }

    // Phase 2: row max of feats (shuffle tree; max has no matmul form).
    float m = wave_max(fmaxf(f0, f1));
    __syncthreads();              // tid0 done reading sh (phase 1)
    if (lane == 0) sh[wave] = m;
    __syncthreads();
    if (tid == 0) {
        float t = sh[0];
#pragma unroll
        for (int w = 1; w < WAVES; ++w) t = fmaxf(t, sh[w]);
        sh_bcast = t;
    }
    __syncthreads();
    m = sh_bcast;

    // Phase 3: Σ exp(f - m) via WMMA reduction -> lse = m + log(sum).
    const float es = wmma_wave_sum(__expf(f0 - m), __expf(f1 - m));
    if (lane == 0) sh[wave] = es;   // safe: last sh reads were before prior sync
    __syncthreads();
    if (tid == 0) {
        float t = 0.0f;
#pragma unroll
        for (int w = 0; w < WAVES; ++w) t += sh[w];
        ws_lse[row] = m + __logf(t);
    }
}

// Kernel B: one block per row. Every block redundantly (and identically)
// reduces the 256 per-row distances -> scalar total, then writes
//   out[row,d] = feats[row,d] - lse[row] + total
__global__ __launch_bounds__(THREADS)
void center_finalize(const float* __restrict__ feats,
                     const float* __restrict__ ws_dist,
                     const float* __restrict__ ws_lse,
                     float* __restrict__ out) {
    const int row  = blockIdx.x;
    const int tid  = threadIdx.x;
    const int wave = tid >> 5;
    const int lane = tid & 31;

    __shared__ float sh[WAVES];
    __shared__ float sh_total;

    // BATCH == THREADS == 256: one ws_dist element per thread.
    const float v = wmma_wave_sum(ws_dist[tid], 0.0f);
    if (lane == 0) sh[wave] = v;
    __syncthreads();
    if (tid == 0) {
        float t = 0.0f;
#pragma unroll
        for (int w = 0; w < WAVES; ++w) t += sh[w];
        sh_total = t;
    }
    __syncthreads();

    const float total = sh_total;
    const float lse   = ws_lse[row];
    const float* f = feats + (size_t)row * FEAT;
    float*       o = out   + (size_t)row * FEAT;
    o[tid]       = f[tid]       - lse + total;
    o[tid + 256] = f[tid + 256] - lse + total;
}

extern "C" void kernel_launch(void* const* d_in, const int* in_sizes, int n_in,
                              void* d_out, int out_size, void* d_ws, size_t ws_size,
                              hipStream_t stream) {
    const float* feats   = (const float*)d_in[0];
    const float* centers = (const float*)d_in[1];
    const int*   labels  = (const int*)d_in[2];

    float* ws_dist = (float*)d_ws;          // 256 floats
    float* ws_lse  = ws_dist + BATCH;       // 256 floats  (total ws use: 2 KB)

    center_rowstats<<<BATCH, THREADS, 0, stream>>>(feats, centers, labels,
                                                   ws_dist, ws_lse);
    center_finalize<<<BATCH, THREADS, 0, stream>>>(feats, ws_dist, ws_lse,
                                                   (float*)d_out);
}